// WaterNet0313_46282567581947
// MI455X (gfx1250) — compile-verified
//
#include <hip/hip_runtime.h>
#include <math.h>

// Problem constants
constexpr int kNF = 6, kNG = 32, kNH = 16, kNR = 15, kHS = 256;
constexpr int kNT = 1000, kNS = 2000;
constexpr int SBP = kHS + 2;           // LDS row pad (bank-conflict avoidance)
constexpr int TCH = 25;                // timesteps per k_km block
constexpr int PGRC = 144 + 256;        // 400 fused pG|pR columns
constexpr int NSH = kNS * kNH;         // 32000 (s,h) streams

typedef float v2f __attribute__((ext_vector_type(2)));
typedef float v8f __attribute__((ext_vector_type(8)));

__device__ __forceinline__ v8f wmma_f32(v2f a, v2f b, v8f c) {
  // D = A(16x4,f32) x B(4x16,f32) + C(16x16,f32)  — exact fp32 path on CDNA5
  return __builtin_amdgcn_wmma_f32_16x16x4_f32(false, a, false, b, (short)0, c,
                                               false, false);
}

// Native CDNA5 V_TANH_F32 (TRANS32). One trailing V_NOP satisfies the
// "1 independent op after TRANS before result use" hazard rule (ISA 5.7.4.3).
__device__ __forceinline__ float fast_tanh(float x) {
#if __has_builtin(__builtin_amdgcn_tanhf)
  return __builtin_amdgcn_tanhf(x);
#elif __has_builtin(__builtin_amdgcn_tanh_f32)
  return __builtin_amdgcn_tanh_f32(x);
#else
  float r;
  asm("v_tanh_f32 %0, %1\n\tv_nop" : "=v"(r) : "v"(x));
  return r;
#endif
}

__device__ __forceinline__ float sigf(float v) { return 1.0f / (1.0f + __expf(-v)); }

// ---------------------------------------------------------------------------
// Kernel 1: state = xc @ W_fc + b_fc ; g = tanh(state); stateb = state + b_kin
// grid (NS/16, HS/16), block 32 (one wave)
// ---------------------------------------------------------------------------
__global__ void __launch_bounds__(32)
k_state(const float* __restrict__ xc, const float* __restrict__ Wfc,
        const float* __restrict__ bfc, const float* __restrict__ bkin,
        float* __restrict__ stateb, float* __restrict__ g) {
  const int lane = threadIdx.x, lm = lane & 15, lh = lane >> 4;
  const int s0 = blockIdx.x * 16, n0 = blockIdx.y * 16;
  v8f acc = {0.f, 0.f, 0.f, 0.f, 0.f, 0.f, 0.f, 0.f};
#pragma unroll
  for (int kk = 0; kk < kNG; kk += 4) {
    v2f a = *(const v2f*)(xc + (size_t)(s0 + lm) * kNG + kk + 2 * lh);
    v2f b;
    b.x = Wfc[(size_t)(kk + 2 * lh) * kHS + n0 + lm];
    b.y = Wfc[(size_t)(kk + 1 + 2 * lh) * kHS + n0 + lm];
    acc = wmma_f32(a, b, acc);
  }
  const int col = n0 + lm;
  const float bb = bfc[col], bk = bkin[col];
#pragma unroll
  for (int r = 0; r < 8; ++r) {
    const int m = r + 8 * lh;
    const float sv = acc[r] + bb;
    g[(size_t)(s0 + m) * kHS + col] = fast_tanh(sv);
    stateb[(size_t)(s0 + m) * kHS + col] = sv + bk;
  }
}

// ---------------------------------------------------------------------------
// Kernel 2: pgr[s, 0:144]   = g @ W_g + b_g
//           pgr[s, 144:400] = g @ W_r + b_r
// grid (NS/16, 25), block 32
// ---------------------------------------------------------------------------
__global__ void __launch_bounds__(32)
k_pgr(const float* __restrict__ g, const float* __restrict__ Wg,
      const float* __restrict__ bg, const float* __restrict__ Wr,
      const float* __restrict__ br, float* __restrict__ pgr) {
  const int lane = threadIdx.x, lm = lane & 15, lh = lane >> 4;
  const int s0 = blockIdx.x * 16;
  const int nt = blockIdx.y;                 // 0..24
  const float* W;
  const float* bias;
  int ncols, col0;
  if (nt < 9) { W = Wg; bias = bg; ncols = 144; col0 = nt * 16; }
  else        { W = Wr; bias = br; ncols = 256; col0 = (nt - 9) * 16; }
  v8f acc = {0.f, 0.f, 0.f, 0.f, 0.f, 0.f, 0.f, 0.f};
#pragma unroll
  for (int kk = 0; kk < kHS; kk += 4) {
    v2f a = *(const v2f*)(g + (size_t)(s0 + lm) * kHS + kk + 2 * lh);
    v2f b;
    b.x = W[(size_t)(kk + 2 * lh) * ncols + col0 + lm];
    b.y = W[(size_t)(kk + 1 + 2 * lh) * ncols + col0 + lm];
    acc = wmma_f32(a, b, acc);
  }
  const float bb = bias[col0 + lm];
  const int outcol = nt * 16 + lm;
#pragma unroll
  for (int r = 0; r < 8; ++r) {
    const int m = r + 8 * lh;
    pgr[(size_t)(s0 + m) * PGRC + outcol] = acc[r] + bb;
  }
}

// ---------------------------------------------------------------------------
// Kernel 3: per-(s,h) parameters. params[p][s*16+h], p: 0..7 scalars, 8..22 W
// ---------------------------------------------------------------------------
__global__ void k_params(const float* __restrict__ pgr, float* __restrict__ params) {
  const int tid = blockIdx.x * blockDim.x + threadIdx.x;
  if (tid >= NSH) return;
  const int s = tid >> 4, h = tid & 15;
  const float* row = pgr + (size_t)s * PGRC;

  const float kp = sigf(row[0 * 16 + h]);
  const float ks = sigf(row[1 * 16 + h]);
  const float kd = sigf(row[2 * 16 + h]);
  const float gd = sigf(row[3 * 16 + h]);
  const float gl0 = sigf(row[4 * 16 + h]) * 10.0f;
  const float gl = gl0 * gl0 * gl0;
  const float qb = fmaxf(row[5 * 16 + h], 0.0f) * 0.1f;
  const float gi = fminf(fmaxf(row[7 * 16 + h] / 6.0f + 0.5f, 0.0f), 1.0f) * 0.5f;
  const float ge = fmaxf(row[8 * 16 + h], 0.0f);

  // ga = softmax over NH of slice 6
  float v6[16], m6 = -1e30f;
#pragma unroll
  for (int j = 0; j < 16; ++j) { v6[j] = row[6 * 16 + j]; m6 = fmaxf(m6, v6[j]); }
  float sum6 = 0.0f;
#pragma unroll
  for (int j = 0; j < 16; ++j) sum6 += __expf(v6[j] - m6);
  const float ga = __expf(v6[h] - m6) / sum6;

  // wR = softmax over NR of pR[h,1:] - (lag - loc)^2
  const float* pr = row + 144 + h * 16;
  const float loc = sigf(pr[0]) * (float)kNR;
  float e[15], mx = -1e30f;
#pragma unroll
  for (int j = 0; j < 15; ++j) {
    const float d = (float)j - loc;
    e[j] = pr[j + 1] - d * d;
    mx = fmaxf(mx, e[j]);
  }
  float sum = 0.0f;
#pragma unroll
  for (int j = 0; j < 15; ++j) { e[j] = __expf(e[j] - mx); sum += e[j]; }
  const float inv = ga / sum;

  params[0 * NSH + tid] = kp;
  params[1 * NSH + tid] = ks;
  params[2 * NSH + tid] = kd;
  params[3 * NSH + tid] = gd;
  params[4 * NSH + tid] = gl;
  params[5 * NSH + tid] = qb;
  params[6 * NSH + tid] = gi;
  params[7 * NSH + tid] = ge;
#pragma unroll
  for (int j = 0; j < 15; ++j) params[(8 + j) * NSH + tid] = e[j] * inv;
}

// ---------------------------------------------------------------------------
// Kernel 4: km[t,s,h] = exp(tanh(f @ W_kin + stateb) @ W_kout + b_kout)
// One wave per (16-site tile, 25-timestep chunk). 80 fp32 WMMAs per t.
// ---------------------------------------------------------------------------
__global__ void __launch_bounds__(32)
k_km(const float* __restrict__ x, const float* __restrict__ stateb,
     const float* __restrict__ Wkin, const float* __restrict__ Wkout,
     const float* __restrict__ bkout, float* __restrict__ km, int t0) {
  __shared__ float sb[16 * SBP];
  __shared__ float hh[16 * SBP];
  const int lane = threadIdx.x, lm = lane & 15, lh = lane >> 4;
  const int s0 = blockIdx.x * 16;

  // stage stateb (state + b_fc + b_kin) into padded LDS
  for (int i = lane; i < 16 * kHS; i += 32) {
    const int r = i >> 8, c = i & 255;
    sb[r * SBP + c] = stateb[(size_t)(s0 + r) * kHS + c];
  }
  // W_kin B-operands: 16 tiles x 2 VGPRs (resident)
  v2f bkin_t[16];
#pragma unroll
  for (int n = 0; n < 16; ++n) {
    bkin_t[n].x = Wkin[(size_t)(2 * lh) * kHS + n * 16 + lm];
    bkin_t[n].y = Wkin[(size_t)(1 + 2 * lh) * kHS + n * 16 + lm];
  }
  // W_kout B-operands: 64 K-steps x 2 VGPRs (resident)
  v2f bko[64];
#pragma unroll
  for (int kk = 0; kk < 64; ++kk) {
    const int k = kk * 4 + 2 * lh;
    bko[kk].x = Wkout[(size_t)k * kNH + lm];
    bko[kk].y = Wkout[(size_t)(k + 1) * kNH + lm];
  }
  const float bk = bkout[lm];
  __syncthreads();

  const int tbase = t0 + blockIdx.y * TCH;
  for (int tt = 0; tt < TCH; ++tt) {
    const int t = tbase + tt;
    // GEMM1: A = f[t, s0..s0+15, 0..3] (x channels 2..5), K=4 exactly
    v2f a1 = *(const v2f*)(x + ((size_t)t * kNS + s0 + lm) * kNF + 2 + 2 * lh);
#pragma unroll
    for (int n = 0; n < 16; ++n) {
      v8f z = {0.f, 0.f, 0.f, 0.f, 0.f, 0.f, 0.f, 0.f};
      v8f d = wmma_f32(a1, bkin_t[n], z);
#pragma unroll
      for (int r = 0; r < 8; ++r) {
        const int m = r + 8 * lh, c = n * 16 + lm;
        hh[m * SBP + c] = fast_tanh(d[r] + sb[m * SBP + c]);
      }
    }
    __syncthreads();
    // GEMM2: pK = H(16x256) @ W_kout(256x16), 64 K-steps of fp32 WMMA
    v8f acc = {0.f, 0.f, 0.f, 0.f, 0.f, 0.f, 0.f, 0.f};
#pragma unroll
    for (int kk = 0; kk < 64; ++kk) {
      v2f a2 = *(const v2f*)(hh + lm * SBP + kk * 4 + 2 * lh);
      acc = wmma_f32(a2, bko[kk], acc);
    }
    const int tl = blockIdx.y * TCH + tt;  // chunk-local t
#pragma unroll
    for (int r = 0; r < 8; ++r) {
      const int m = r + 8 * lh;
      km[((size_t)tl * kNS + s0 + m) * kNH + lm] = __expf(acc[r] + bk);
    }
    __syncthreads();
  }
}

// ---------------------------------------------------------------------------
// Kernel 5: sequential bucket scan + fused 15-tap routing convolution.
// One lane per (s,h). h-reduction via shfl_xor within 16-lane groups (wave32).
// ---------------------------------------------------------------------------
__global__ void k_scan(const float* __restrict__ x, const float* __restrict__ km,
                       const float* __restrict__ params, float* __restrict__ Sf_g,
                       float* __restrict__ Ss_g, float* __restrict__ Sd_g,
                       float* __restrict__ ring_g, float* __restrict__ out,
                       int t0, int tc, int first) {
  const int tid = blockIdx.x * blockDim.x + threadIdx.x;
  if (tid >= NSH) return;
  const int s = tid >> 4, h = tid & 15;

  const float kp = params[0 * NSH + tid];
  const float ks = params[1 * NSH + tid];
  const float kd = params[2 * NSH + tid];
  const float gd = params[3 * NSH + tid];
  const float gl = params[4 * NSH + tid];
  const float qb = params[5 * NSH + tid];
  const float gi = params[6 * NSH + tid];
  const float ge = params[7 * NSH + tid];
  float Wl[15];
#pragma unroll
  for (int j = 0; j < 15; ++j) Wl[j] = params[(8 + j) * NSH + tid];

  float Sf, Ss, Sd, acc[15];
  if (first) {
    Sf = Ss = Sd = 0.f;
#pragma unroll
    for (int j = 0; j < 15; ++j) acc[j] = 0.f;
  } else {
    Sf = Sf_g[tid]; Ss = Ss_g[tid]; Sd = Sd_g[tid];
#pragma unroll
    for (int j = 0; j < 15; ++j) acc[j] = ring_g[(size_t)j * NSH + tid];
  }

  for (int tt = 0; tt < tc; ++tt) {
    const int t = t0 + tt;
    const float* xb = x + ((size_t)t * kNS + s) * kNF;
    v2f pe  = *(const v2f*)(xb);       // Prcp, Evp
    v2f t12 = *(const v2f*)(xb + 2);   // T1, T2
    const float kmt = km[((size_t)tt * kNS + s) * kNH + h];

    const float vL = fminf(fmaxf(t12.y / fmaxf(t12.y - t12.x, 1e-6f), 0.f), 1.f);
    const float Pl = pe.x * vL, Ps = pe.x * (1.f - vL), Ev = pe.y;

    const float Sf1 = Sf + Ps;
    const float qf  = fminf(Sf1, kmt);
    Sf = Sf1 - qf;
    const float Pin = Pl + qf;
    const float H   = fmaxf(Ss + Pin * (1.f - gi) - Ev * ge, 0.f);
    const float qp  = fmaxf(H - gl, 0.f) * kp;
    const float qo  = fminf(H, gl) * ks;
    Ss = H - qp - qo;
    const float D  = Sd + Pin * gi + qo * gd;
    const float qd = D * kd + qb;
    Sd = D * (1.f - kd);
    const float q = qp + qo * (1.f - gd) + qd;

#pragma unroll
    for (int j = 0; j < 15; ++j) acc[j] += q * Wl[j];

    if (t >= kNR - 1) {                 // row (t-14) complete: reduce over h
      float r = acc[14];
      r += __shfl_xor(r, 1, 16);
      r += __shfl_xor(r, 2, 16);
      r += __shfl_xor(r, 4, 16);
      r += __shfl_xor(r, 8, 16);
      if (h == 0) out[(size_t)(t - (kNR - 1)) * kNS + s] = r;
    }
#pragma unroll
    for (int j = 14; j >= 1; --j) acc[j] = acc[j - 1];
    acc[0] = 0.f;
  }

  Sf_g[tid] = Sf; Ss_g[tid] = Ss; Sd_g[tid] = Sd;
#pragma unroll
  for (int j = 0; j < 15; ++j) ring_g[(size_t)j * NSH + tid] = acc[j];
}

// ---------------------------------------------------------------------------
extern "C" void kernel_launch(void* const* d_in, const int* in_sizes, int n_in,
                              void* d_out, int out_size, void* d_ws, size_t ws_size,
                              hipStream_t stream) {
  (void)in_sizes; (void)n_in; (void)out_size;
  const float* x     = (const float*)d_in[0];
  const float* xc    = (const float*)d_in[1];
  const float* W_fc  = (const float*)d_in[2];
  const float* b_fc  = (const float*)d_in[3];
  const float* W_r   = (const float*)d_in[4];
  const float* b_r   = (const float*)d_in[5];
  const float* W_g   = (const float*)d_in[6];
  const float* b_g   = (const float*)d_in[7];
  const float* W_kin = (const float*)d_in[8];
  const float* b_kin = (const float*)d_in[9];
  const float* W_kout= (const float*)d_in[10];
  const float* b_kout= (const float*)d_in[11];
  float* out = (float*)d_out;

  char* p = (char*)d_ws;
  auto alloc = [&](size_t bytes) {
    float* r = (float*)p;
    p += (bytes + 255) & ~(size_t)255;
    return r;
  };
  float* stateb = alloc((size_t)kNS * kHS * 4);
  float* g      = alloc((size_t)kNS * kHS * 4);
  float* pgr    = alloc((size_t)kNS * PGRC * 4);
  float* params = alloc((size_t)23 * NSH * 4);
  float* Sf     = alloc((size_t)NSH * 4);
  float* Ss     = alloc((size_t)NSH * 4);
  float* Sd     = alloc((size_t)NSH * 4);
  float* ring   = alloc((size_t)15 * NSH * 4);
  float* kmbuf  = (float*)p;

  const size_t used  = (size_t)(p - (char*)d_ws);
  const size_t per_t = (size_t)kNS * kNH * 4;   // 128 KB per timestep
  long avail = (ws_size > used) ? (long)(ws_size - used) : 0;
  int Tc = (int)(avail / (long)per_t);
  Tc = (Tc / TCH) * TCH;
  if (Tc < TCH) Tc = TCH;       // minimum viable chunk
  if (Tc > kNT) Tc = kNT;

  k_state<<<dim3(kNS / 16, kHS / 16), 32, 0, stream>>>(xc, W_fc, b_fc, b_kin,
                                                       stateb, g);
  k_pgr<<<dim3(kNS / 16, 25), 32, 0, stream>>>(g, W_g, b_g, W_r, b_r, pgr);
  k_params<<<NSH / 256, 256, 0, stream>>>(pgr, params);

  int first = 1;
  for (int t0 = 0; t0 < kNT; t0 += Tc) {
    const int tc = (kNT - t0 < Tc) ? (kNT - t0) : Tc;   // multiple of 25
    k_km<<<dim3(kNS / 16, tc / TCH), 32, 0, stream>>>(x, stateb, W_kin, W_kout,
                                                      b_kout, kmbuf, t0);
    k_scan<<<NSH / 256, 256, 0, stream>>>(x, kmbuf, params, Sf, Ss, Sd, ring,
                                          out, t0, tc, first);
    first = 0;
  }
}